// EdgeAttention_85676007620986
// MI455X (gfx1250) — compile-verified
//
#include <hip/hip_runtime.h>
#include <math.h>
#include <stdint.h>

// ---------------------------------------------------------------------------
// EdgeAttention for MI455X (gfx1250, wave32, WMMA + async-to-LDS).
//
// Key simplification: KNN_K == 1  =>  softmax over a single element == 1
//   => attention output == v == Wv @ (x_j - x_n).  Wq/Wk are dead.
//   Wc@x + Wv@(x_j - x_n) == Cx + Vx[j] - Vx[n]  with Vx=Wv@x, Cx=Wc@x.
//
// Pipeline (all fp32, exact-precision WMMA f32 16x16x4):
//   sq      : per-point squared norms
//   knn     : Gram matrix via WMMA + fused row-argmin (34.4 GFLOP, dominant).
//             Column panels double-buffered in LDS via
//             global_load_async_to_lds_b128 (ASYNCcnt), overlapping DMA with
//             WMMA; A fragments register-resident.
//   proj    : Vx = Wv@x, S1 = Wc@x            (WMMA)
//   gather  : S1 += Vx[j] - Vx[n]
//   stats1  : BN1 mean/rstd (deterministic, no atomics)
//   ffn     : h=BN1(S1); S2 = h + W2@lrelu(W1@h)  fused per 64-col tile (WMMA)
//   stats2  : BN2 mean/rstd
//   norm    : out = BN2(S2)
//
// Workspace: idx(128K) | sq(128K) | ms(2K) | Vx@1MB(16MB) | S1@17MB | S2@33MB
// ---------------------------------------------------------------------------

#define B_SZ   8
#define C_DIM  128
#define N_PTS  4096
#define O_DIM  128

#define KP   132   // padded stride for [point][channel] LDS tiles
#define LDB  136   // padded stride for K-major async panels (544B rows: 16B
                   // aligned for B128 writes; 8k-mod-64 bank rotation)
#define LDU  260   // padded stride for 256-wide FFN intermediate

typedef float v2f __attribute__((ext_vector_type(2)));
typedef float v8f __attribute__((ext_vector_type(8)));

// One wave computes a 16x16 f32 tile. A and B stored point-major: A[m][k],
// B[n][k]. Fragment layout per CDNA5 ISA 7.12.2 (32-bit A 16x4): lanes 0-15
// hold K=0,1; lanes 16-31 hold K=2,3; M (resp. N) = lane & 15.
__device__ __forceinline__ v8f wmma_tile(const float* __restrict__ A, int ldA,
                                         const float* __restrict__ B, int ldB,
                                         int K) {
  const int lane = threadIdx.x & 31;
  const int lm = lane & 15;
  const int kb = (lane >> 4) << 1;
  v8f acc = {0.f, 0.f, 0.f, 0.f, 0.f, 0.f, 0.f, 0.f};
#pragma unroll
  for (int k = 0; k < K; k += 4) {
    v2f a;
    a.x = A[lm * ldA + k + kb];
    a.y = A[lm * ldA + k + kb + 1];
    v2f bb;
    bb.x = B[lm * ldB + k + kb];
    bb.y = B[lm * ldB + k + kb + 1];
    acc = __builtin_amdgcn_wmma_f32_16x16x4_f32(false, a, false, bb,
                                                (short)0, acc, false, false);
  }
  return acc;
}

// Variant with B stored K-major: B[k][n] (stride ldB). The {k,k+1} dword pair
// is 4*ldB bytes apart -> folds into ds_load_2addr_b32.
__device__ __forceinline__ v8f wmma_tile_kmaj(const float* __restrict__ A, int ldA,
                                              const float* __restrict__ B, int ldB,
                                              int K) {
  const int lane = threadIdx.x & 31;
  const int lm = lane & 15;
  const int kb = (lane >> 4) << 1;
  v8f acc = {0.f, 0.f, 0.f, 0.f, 0.f, 0.f, 0.f, 0.f};
#pragma unroll
  for (int k = 0; k < K; k += 4) {
    v2f a;
    a.x = A[lm * ldA + k + kb];
    a.y = A[lm * ldA + k + kb + 1];
    v2f bb;
    bb.x = B[(k + kb) * ldB + lm];
    bb.y = B[(k + kb + 1) * ldB + lm];
    acc = __builtin_amdgcn_wmma_f32_16x16x4_f32(false, a, false, bb,
                                                (short)0, acc, false, false);
  }
  return acc;
}

// D tile layout (ISA 7.12.2): element (M = r + 8*(lane>>4), N = lane&15) in acc[r].
__device__ __forceinline__ void store_tile(float* dst, int ldD, v8f acc) {
  const int lane = threadIdx.x & 31;
  const int lm = lane & 15;
  const int hi = lane >> 4;
#pragma unroll
  for (int r = 0; r < 8; ++r) dst[(r + 8 * hi) * ldD + lm] = acc[r];
}

// Async DMA: memory -> LDS, 16B per lane, no VGPR data, tracked by ASYNCcnt.
// VDST operand is the wave-relative LDS byte address = low 32 bits of the
// generic pointer (aperture layout keeps the LDS offset in addr[31:0]).
__device__ __forceinline__ void async_copy_b128(const float* g, const float* l) {
  asm volatile("global_load_async_to_lds_b128 %0, %1, off"
               :: "v"((uint32_t)(uintptr_t)l),
                  "v"((uint64_t)(uintptr_t)g)
               : "memory");
}

__device__ __forceinline__ void wait_async0() {
  asm volatile("s_wait_asynccnt 0x0" ::: "memory");
}

// Stage one 128-col panel (K-major [c][n], stride LDB) with async DMA.
// Wave owns 32 rows; one b128 per row covers 32 lanes * 16B = 512B = full row.
__device__ __forceinline__ void stage_panel_async(const float* __restrict__ xb,
                                                  int colBase, float* lbuf,
                                                  int wave, int lane) {
#pragma unroll
  for (int i = 0; i < 32; ++i) {
    const int c = wave * 32 + i;
    async_copy_b128(xb + (size_t)c * N_PTS + colBase + lane * 4,
                    lbuf + c * LDB + lane * 4);
  }
}

// ---------------------------------------------------------------- sq norms --
__global__ __launch_bounds__(256) void sq_kernel(const float* __restrict__ x,
                                                 float* __restrict__ sq) {
  const int b = blockIdx.y;
  const int n = blockIdx.x * 256 + threadIdx.x;
  const float* xb = x + (size_t)b * C_DIM * N_PTS;
  float s = 0.f;
#pragma unroll 8
  for (int c = 0; c < C_DIM; ++c) {
    float v = xb[(size_t)c * N_PTS + n];
    s += v * v;
  }
  sq[(size_t)b * N_PTS + n] = s;
}

// ------------------------------------------------------- KNN via WMMA Gram --
// Block: 128 threads (4 waves). Block owns 64 rows; each wave 16 rows.
// Column panels of 128 double-buffered in LDS via async DMA.
__global__ __launch_bounds__(128) void knn_kernel(const float* __restrict__ x,
                                                  const float* __restrict__ sq,
                                                  int* __restrict__ idxOut) {
  const int b = blockIdx.y;
  const int rowBase = blockIdx.x * 64;
  const float* xb = x + (size_t)b * C_DIM * N_PTS;
  const float* sqb = sq + (size_t)b * N_PTS;

  __shared__ float As[64 * KP];                       // A rows, [point][c]
  __shared__ __align__(16) float Bs[2 * 128 * LDB];   // panels, K-major [c][n]

  const int tid = threadIdx.x;
  const int wave = tid >> 5;
  const int lane = tid & 31;
  const int lm = lane & 15;
  const int hi = lane >> 4;

  // Kick off panel 0 DMA immediately, then stage A rows (transposed) while
  // the async unit fills Bs.
  stage_panel_async(xb, 0, Bs, wave, lane);

  for (int c = wave; c < C_DIM; c += 4) {
    float2 v = *(const float2*)(xb + (size_t)c * N_PTS + rowBase + lane * 2);
    As[(lane * 2 + 0) * KP + c] = v.x;
    As[(lane * 2 + 1) * KP + c] = v.y;
  }

  float bestD[8];
  int bestI[8];
  float rowSq[8];
#pragma unroll
  for (int r = 0; r < 8; ++r) {
    rowSq[r] = sqb[rowBase + wave * 16 + r + 8 * hi];
    bestD[r] = 3.4e38f;
    bestI[r] = 0;
  }

  const float* Arow = As + wave * 16 * KP;
  const int NPANEL = N_PTS / 128;

  for (int p = 0; p < NPANEL; ++p) {
    wait_async0();       // panel p resident (in-order completion)
    __syncthreads();     // all waves waited; prior compute on other buf done
    if (p + 1 < NPANEL)  // DMA panel p+1 underneath this panel's WMMA work
      stage_panel_async(xb, (p + 1) * 128, Bs + ((p + 1) & 1) * (128 * LDB),
                        wave, lane);

    const float* Bbuf = Bs + (p & 1) * (128 * LDB);
    const int colBase = p * 128;
    for (int nt = 0; nt < 8; ++nt) {
      v8f g = wmma_tile_kmaj(Arow, KP, Bbuf + nt * 16, LDB, C_DIM);
      const int ncol = colBase + nt * 16 + lm;
      const float colSq = sqb[ncol];
#pragma unroll
      for (int r = 0; r < 8; ++r) {
        const int mrow = rowBase + wave * 16 + r + 8 * hi;
        float d = rowSq[r] + colSq - 2.0f * g[r];
        if (mrow == ncol) d = 3.4e38f;  // exclude self (top_k drops it)
        int ci = ncol;
        // butterfly argmin across the 16 lanes of this half-wave
#pragma unroll
        for (int m = 1; m < 16; m <<= 1) {
          float od = __shfl_xor(d, m, 32);
          int oi = __shfl_xor(ci, m, 32);
          if (od < d || (od == d && oi < ci)) { d = od; ci = oi; }
        }
        if (d < bestD[r] || (d == bestD[r] && ci < bestI[r])) {
          bestD[r] = d;
          bestI[r] = ci;
        }
      }
    }
  }

  if (lm == 0) {
#pragma unroll
    for (int r = 0; r < 8; ++r)
      idxOut[(size_t)b * N_PTS + rowBase + wave * 16 + r + 8 * hi] = bestI[r];
  }
}

// ------------------------------------------------ Vx = Wv@x ; S1 = Wc@x -----
__global__ __launch_bounds__(128) void proj_kernel(const float* __restrict__ x,
                                                   const float* __restrict__ Wv,
                                                   const float* __restrict__ Wc,
                                                   float* __restrict__ Vx,
                                                   float* __restrict__ S1) {
  const int b = blockIdx.y;
  const int nBase = blockIdx.x * 64;
  const float* xb = x + (size_t)b * C_DIM * N_PTS;

  __shared__ float xs[64 * KP];
  const int wave = threadIdx.x >> 5;
  const int lane = threadIdx.x & 31;

  for (int c = wave; c < C_DIM; c += 4) {
    float2 v = *(const float2*)(xb + (size_t)c * N_PTS + nBase + lane * 2);
    xs[(lane * 2 + 0) * KP + c] = v.x;
    xs[(lane * 2 + 1) * KP + c] = v.y;
  }
  __syncthreads();

  const float* Bn = xs + wave * 16 * KP;  // this wave's 16-column strip
  for (int pass = 0; pass < 2; ++pass) {
    const float* W = pass ? Wc : Wv;     // weight A-fragments straight from L2
    float* Out = pass ? S1 : Vx;
    for (int mt = 0; mt < 8; ++mt) {
      v8f acc = wmma_tile(W + mt * 16 * C_DIM, C_DIM, Bn, KP, C_DIM);
      store_tile(Out + ((size_t)b * O_DIM + mt * 16) * N_PTS + nBase + wave * 16,
                 N_PTS, acc);
    }
  }
}

// ------------------------------------- S1 += Vx[:, j(n)] - Vx[:, n] ---------
__global__ __launch_bounds__(256) void gather_kernel(const float* __restrict__ Vx,
                                                     const int* __restrict__ idx,
                                                     float* __restrict__ S1) {
  const int b = blockIdx.y;
  const int n = blockIdx.x * 256 + threadIdx.x;
  const int j = idx[(size_t)b * N_PTS + n];
  const float* V = Vx + (size_t)b * O_DIM * N_PTS;
  float* S = S1 + (size_t)b * O_DIM * N_PTS;
#pragma unroll 4
  for (int o = 0; o < O_DIM; ++o)
    S[(size_t)o * N_PTS + n] += V[(size_t)o * N_PTS + j] - V[(size_t)o * N_PTS + n];
}

// --------------------------- BN stats: one block per channel (deterministic) -
__global__ __launch_bounds__(256) void stats_kernel(const float* __restrict__ S,
                                                    float* __restrict__ ms) {
  const int o = blockIdx.x;
  float s = 0.f, q = 0.f;
  for (int b = 0; b < B_SZ; ++b) {
    const float* p = S + ((size_t)b * O_DIM + o) * N_PTS;
    for (int n = threadIdx.x; n < N_PTS; n += 256) {
      float v = p[n];
      s += v;
      q += v * v;
    }
  }
#pragma unroll
  for (int m = 1; m < 32; m <<= 1) {
    s += __shfl_xor(s, m, 32);
    q += __shfl_xor(q, m, 32);
  }
  __shared__ float ls[8], lq[8];
  const int wave = threadIdx.x >> 5, lane = threadIdx.x & 31;
  if (lane == 0) { ls[wave] = s; lq[wave] = q; }
  __syncthreads();
  if (threadIdx.x == 0) {
    s = 0.f; q = 0.f;
    for (int w = 0; w < 8; ++w) { s += ls[w]; q += lq[w]; }
    const float inv = 1.0f / (float)(B_SZ * N_PTS);
    float mean = s * inv;
    float var = q * inv - mean * mean;
    ms[o] = mean;
    ms[128 + o] = rsqrtf(var + 1e-5f);
  }
}

// ---------------- fused FFN: h = BN1(S1); S2 = h + W2 @ lrelu(W1 @ h) -------
__global__ __launch_bounds__(128) void ffn_kernel(const float* __restrict__ S1,
                                                  const float* __restrict__ ms1,
                                                  const float* __restrict__ g1,
                                                  const float* __restrict__ be1,
                                                  const float* __restrict__ W1,
                                                  const float* __restrict__ W2,
                                                  float* __restrict__ S2) {
  const int b = blockIdx.y;
  const int nBase = blockIdx.x * 64;
  __shared__ float Hs[64 * KP];   // normalized h, [n][o]
  __shared__ float Us[64 * LDU];  // lrelu(W1@h),  [n][o2], o2 in [0,256)

  const int wave = threadIdx.x >> 5;
  const int lane = threadIdx.x & 31;
  const int lm = lane & 15;
  const int hi = lane >> 4;

  for (int o = wave; o < O_DIM; o += 4) {
    const float mean = ms1[o], rstd = ms1[128 + o];
    const float sc = g1[o] * rstd;
    const float off = be1[o] - mean * sc;
    const float* src = S1 + ((size_t)b * O_DIM + o) * N_PTS + nBase;
    float2 v = *(const float2*)(src + lane * 2);
    Hs[(lane * 2 + 0) * KP + o] = v.x * sc + off;
    Hs[(lane * 2 + 1) * KP + o] = v.y * sc + off;
  }
  __syncthreads();

  const float* Bn = Hs + wave * 16 * KP;
  for (int mt = 0; mt < 16; ++mt) {
    v8f acc = wmma_tile(W1 + mt * 16 * C_DIM, C_DIM, Bn, KP, C_DIM);
#pragma unroll
    for (int r = 0; r < 8; ++r) {
      float v = acc[r];
      v = (v > 0.f) ? v : 0.2f * v;  // leaky_relu(., 0.2)
      Us[(wave * 16 + lm) * LDU + mt * 16 + r + 8 * hi] = v;
    }
  }
  __syncthreads();

  const float* Un = Us + wave * 16 * LDU;
  for (int mt = 0; mt < 8; ++mt) {
    v8f acc = wmma_tile(W2 + mt * 16 * 256, 256, Un, LDU, 256);
#pragma unroll
    for (int r = 0; r < 8; ++r) {
      const int o = mt * 16 + r + 8 * hi;
      const int n = wave * 16 + lm;
      S2[((size_t)b * O_DIM + o) * N_PTS + nBase + n] = acc[r] + Hs[n * KP + o];
    }
  }
}

// -------------------------------------------------- out = BN2(S2) ----------
__global__ __launch_bounds__(256) void norm_kernel(const float* __restrict__ S2,
                                                   const float* __restrict__ ms2,
                                                   const float* __restrict__ g2,
                                                   const float* __restrict__ be2,
                                                   float* __restrict__ out) {
  const int o = blockIdx.x & 127;
  const int b = blockIdx.x >> 7;
  const float mean = ms2[o], rstd = ms2[128 + o];
  const float sc = g2[o] * rstd;
  const float off = be2[o] - mean * sc;
  const float4* p = (const float4*)(S2 + ((size_t)b * O_DIM + o) * N_PTS);
  float4* q = (float4*)(out + ((size_t)b * O_DIM + o) * N_PTS);
  for (int i = threadIdx.x; i < N_PTS / 4; i += 256) {
    float4 v = p[i];
    v.x = v.x * sc + off;
    v.y = v.y * sc + off;
    v.z = v.z * sc + off;
    v.w = v.w * sc + off;
    q[i] = v;
  }
}

// ---------------------------------------------------------------------------
extern "C" void kernel_launch(void* const* d_in, const int* in_sizes, int n_in,
                              void* d_out, int out_size, void* d_ws, size_t ws_size,
                              hipStream_t stream) {
  (void)in_sizes; (void)n_in; (void)out_size; (void)ws_size;
  const float* x  = (const float*)d_in[0];
  // d_in[1]=Wq, d_in[2]=Wk: dead (softmax over K=1 is identically 1).
  const float* Wv = (const float*)d_in[3];
  const float* Wc = (const float*)d_in[4];
  const float* W1 = (const float*)d_in[5];
  const float* W2 = (const float*)d_in[6];
  const float* g1 = (const float*)d_in[7];
  const float* b1 = (const float*)d_in[8];
  const float* g2 = (const float*)d_in[9];
  const float* b2 = (const float*)d_in[10];
  float* out = (float*)d_out;

  char* ws = (char*)d_ws;
  int*   idx = (int*)(ws);
  float* sq  = (float*)(ws + (size_t)(128 << 10));
  float* ms1 = (float*)(ws + (size_t)(256 << 10));
  float* ms2 = ms1 + 256;
  float* Vx  = (float*)(ws + ((size_t)1 << 20));
  float* S1  = (float*)(ws + ((size_t)17 << 20));
  float* S2  = (float*)(ws + ((size_t)33 << 20));

  sq_kernel<<<dim3(N_PTS / 256, B_SZ), 256, 0, stream>>>(x, sq);
  knn_kernel<<<dim3(N_PTS / 64, B_SZ), 128, 0, stream>>>(x, sq, idx);
  proj_kernel<<<dim3(N_PTS / 64, B_SZ), 128, 0, stream>>>(x, Wv, Wc, Vx, S1);
  gather_kernel<<<dim3(N_PTS / 256, B_SZ), 256, 0, stream>>>(Vx, idx, S1);
  stats_kernel<<<dim3(O_DIM), 256, 0, stream>>>(S1, ms1);
  ffn_kernel<<<dim3(N_PTS / 64, B_SZ), 128, 0, stream>>>(S1, ms1, g1, b1, W1, W2, S2);
  stats_kernel<<<dim3(O_DIM), 256, 0, stream>>>(S2, ms2);
  norm_kernel<<<dim3(B_SZ * O_DIM), 256, 0, stream>>>(S2, ms2, g2, b2, out);
}